// HeteroGeneralLayer_32667521253698
// MI455X (gfx1250) — compile-verified
//
#include <hip/hip_runtime.h>

typedef __attribute__((ext_vector_type(2))) float v2f;
typedef __attribute__((ext_vector_type(8))) float v8f;

__device__ __forceinline__ float wave_reduce_sum(float v) {
  #pragma unroll
  for (int off = 16; off > 0; off >>= 1) v += __shfl_xor(v, off, 32);
  return v;
}

// ---------------------------------------------------------------- degrees
__global__ __launch_bounds__(256)
void k_degrees(const int* __restrict__ src, const int* __restrict__ dst,
               float* __restrict__ degS, float* __restrict__ degD, int E) {
  int i = blockIdx.x * blockDim.x + threadIdx.x;
  if (i < E) {
    unsafeAtomicAdd(&degS[src[i]], 1.0f);
    unsafeAtomicAdd(&degD[dst[i]], 1.0f);
  }
}

// ------------------------------------------- m = (A * rsqrt(degOut)) @ W
// One wave computes a full 16x128 row block: 8 accumulators, A-fragment
// loaded once per k-step and reused for 8 back-to-back WMMAs.
// A-frag: lane L -> row=L%16, a = A[row][k0+2*(L/16) .. +1] (float2).
// B-frag: lane L -> col=L%16, b = W[k0+2*(L/16)][col], W[k0+2*(L/16)+1][col].
// C/D:    VGPR i -> row = i + 8*(L/16), col = L%16.
__global__ __launch_bounds__(256)
void k_gemm_scaled(const float* __restrict__ A, const float* __restrict__ degOut,
                   const float* __restrict__ W, float* __restrict__ Out, int N) {
  const int lane = threadIdx.x & 31;
  const int rowTile = blockIdx.x * 8 + (threadIdx.x >> 5);   // wave per 16-row block
  if (rowTile >= (N >> 4)) return;          // wave-uniform guard (EXEC stays all-1s)
  const int rowBase = rowTile << 4;
  const int r = lane & 15;
  const int h = lane >> 4;

  const int arow = rowBase + r;
  const float rs = rsqrtf(fmaxf(degOut[arow], 1.0f));
  const float* __restrict__ Arow = A + (size_t)arow * 128;

  v8f acc[8] = {};
  for (int k0 = 0; k0 < 128; k0 += 4) {
    const int ka = k0 + 2 * h;
    v2f a;
    a.x = Arow[ka] * rs;
    a.y = Arow[ka + 1] * rs;
    const float* __restrict__ Wk0 = W + (size_t)ka * 128 + r;
    const float* __restrict__ Wk1 = Wk0 + 128;
    #pragma unroll
    for (int t = 0; t < 8; ++t) {
      v2f b;
      b.x = Wk0[t * 16];
      b.y = Wk1[t * 16];
      acc[t] = __builtin_amdgcn_wmma_f32_16x16x4_f32(false, a, false, b, (short)0,
                                                     acc[t], false, false);
    }
  }
  #pragma unroll
  for (int t = 0; t < 8; ++t) {
    float* __restrict__ o = Out + (size_t)(rowBase + 8 * h) * 128 + t * 16 + r;
    #pragma unroll
    for (int i = 0; i < 8; ++i) o[(size_t)i * 128] = acc[t][i];
  }
}

// ------------------------------------------------ agg[dst] += m[src]  (wave per edge)
__global__ __launch_bounds__(256)
void k_scatter(const float* __restrict__ M, const int* __restrict__ src,
               const int* __restrict__ dst, float* __restrict__ Z, int E) {
  const int wave = (blockIdx.x * blockDim.x + threadIdx.x) >> 5;
  const int lane = threadIdx.x & 31;
  if (wave >= E) return;
  const int s = src[wave];
  const int d = dst[wave];
  const float4 v = *(const float4*)(M + (size_t)s * 128 + lane * 4);
  float* o = Z + (size_t)d * 128 + lane * 4;
  unsafeAtomicAdd(o + 0, v.x);
  unsafeAtomicAdd(o + 1, v.y);
  unsafeAtomicAdd(o + 2, v.z);
  unsafeAtomicAdd(o + 3, v.w);
}

// ------------------------------------------------ z *= rsqrt(max(degIn,1))
__global__ __launch_bounds__(256)
void k_scale_rows(float* __restrict__ Z, const float* __restrict__ degIn, int N) {
  const int i = blockIdx.x * blockDim.x + threadIdx.x;   // one float4 per thread
  if (i >= N * 32) return;
  const int row = i >> 5;
  const float rs = rsqrtf(fmaxf(degIn[row], 1.0f));
  float4* p = (float4*)Z + i;
  float4 v = *p;
  v.x *= rs; v.y *= rs; v.z *= rs; v.w *= rs;
  *p = v;
}

// ----------------- wAcc += sum_rows( tanh(Z@attW + b) . q )  via WMMA row blocks
__global__ __launch_bounds__(256)
void k_attention(const float* __restrict__ Z, const float* __restrict__ attW,
                 const float* __restrict__ attB, const float* __restrict__ attQ,
                 float* __restrict__ wAcc, int N) {
  const int lane = threadIdx.x & 31;
  const int rowTile = blockIdx.x * 8 + (threadIdx.x >> 5);
  if (rowTile >= (N >> 4)) return;
  const int rowBase = rowTile << 4;
  const int r = lane & 15;
  const int h = lane >> 4;
  const float* __restrict__ Zrow = Z + (size_t)(rowBase + r) * 128;

  v8f acc[8] = {};
  for (int k0 = 0; k0 < 128; k0 += 4) {
    const int ka = k0 + 2 * h;
    v2f a;
    a.x = Zrow[ka];
    a.y = Zrow[ka + 1];
    const float* __restrict__ Wk0 = attW + (size_t)ka * 128 + r;
    const float* __restrict__ Wk1 = Wk0 + 128;
    #pragma unroll
    for (int t = 0; t < 8; ++t) {
      v2f b;
      b.x = Wk0[t * 16];
      b.y = Wk1[t * 16];
      acc[t] = __builtin_amdgcn_wmma_f32_16x16x4_f32(false, a, false, b, (short)0,
                                                     acc[t], false, false);
    }
  }
  float part = 0.0f;
  #pragma unroll
  for (int t = 0; t < 8; ++t) {
    const int col = t * 16 + r;
    const float bcol = attB[col];
    const float qcol = attQ[col];
    #pragma unroll
    for (int i = 0; i < 8; ++i) part += tanhf(acc[t][i] + bcol) * qcol;
  }
  part = wave_reduce_sum(part);
  if (lane == 0) unsafeAtomicAdd(wAcc, part);
}

// ------------------------------------------------ beta = softmax(w/N) (R=2), +consts
__global__ void k_softmax2(const float* __restrict__ wAcc, float* __restrict__ beta,
                           float invN) {
  if (threadIdx.x == 0) {
    const float w0 = wAcc[0] * invN;
    const float w1 = wAcc[1] * invN;
    const float m = fmaxf(w0, w1);
    const float e0 = expf(w0 - m);
    const float e1 = expf(w1 - m);
    const float inv = 1.0f / (e0 + e1);
    beta[0] = e0 * inv;
    beta[1] = e1 * inv;
    beta[2] = 1.0f;   // authors: softmax over single relation
    beta[3] = 0.0f;
  }
}

// -------------------------- column sum / sumsq of x = b0*Z1 + b1*Z2 (coalesced)
__global__ __launch_bounds__(128)
void k_colstats(const float* __restrict__ Z1, const float* __restrict__ Z2,
                const float* __restrict__ beta, int i0, int i1,
                float* __restrict__ stats, int N, int rowsPerBlock) {
  const int col = threadIdx.x;              // blockDim = 128
  const float b0 = beta[i0];
  const float b1 = beta[i1];
  const int r0 = blockIdx.x * rowsPerBlock;
  const int r1 = min(r0 + rowsPerBlock, N);
  float s = 0.0f, s2 = 0.0f;
  for (int row = r0; row < r1; ++row) {
    const float x = b0 * Z1[(size_t)row * 128 + col] + b1 * Z2[(size_t)row * 128 + col];
    s += x;
    s2 += x * x;
  }
  unsafeAtomicAdd(&stats[col], s);
  unsafeAtomicAdd(&stats[128 + col], s2);
}

// ------------------------------------------------ mu / invstd per column
__global__ void k_colnorm(const float* __restrict__ stats, float* __restrict__ cn,
                          float invN) {
  const int c = threadIdx.x;                // 128 threads
  const float mu = stats[c] * invN;
  const float var = stats[128 + c] * invN - mu * mu;
  cn[c] = mu;
  cn[128 + c] = rsqrtf(var + 1e-5f);
}

// ----------- fused: x = b0*Z1+b1*Z2 -> BatchNorm -> row L2 normalize -> out
__global__ __launch_bounds__(256)
void k_finalize(const float* __restrict__ Z1, const float* __restrict__ Z2,
                const float* __restrict__ beta, int i0, int i1,
                const float* __restrict__ cn,
                const float* __restrict__ gamma, const float* __restrict__ bnbeta,
                float* __restrict__ out, int N) {
  const int wave = (blockIdx.x * blockDim.x + threadIdx.x) >> 5;  // one row per wave
  const int lane = threadIdx.x & 31;
  if (wave >= N) return;
  const int c = lane * 4;
  const float b0 = beta[i0];
  const float b1 = beta[i1];
  const float4 v1 = *(const float4*)(Z1 + (size_t)wave * 128 + c);
  const float4 v2 = *(const float4*)(Z2 + (size_t)wave * 128 + c);
  const float raw[4] = {b0 * v1.x + b1 * v2.x, b0 * v1.y + b1 * v2.y,
                        b0 * v1.z + b1 * v2.z, b0 * v1.w + b1 * v2.w};
  float xn[4];
  float ss = 0.0f;
  #pragma unroll
  for (int j = 0; j < 4; ++j) {
    const float t = (raw[j] - cn[c + j]) * cn[128 + c + j] * gamma[c + j] + bnbeta[c + j];
    xn[j] = t;
    ss += t * t;
  }
  ss = wave_reduce_sum(ss);
  const float sc = rsqrtf(ss + 1e-12f);
  float4 o;
  o.x = xn[0] * sc; o.y = xn[1] * sc; o.z = xn[2] * sc; o.w = xn[3] * sc;
  *(float4*)(out + (size_t)wave * 128 + c) = o;
}

extern "C" void kernel_launch(void* const* d_in, const int* in_sizes, int n_in,
                              void* d_out, int out_size, void* d_ws, size_t ws_size,
                              hipStream_t stream) {
  const float* h_paper  = (const float*)d_in[0];
  const float* h_author = (const float*)d_in[1];
  const float* W_r1     = (const float*)d_in[2];
  const float* W_r2     = (const float*)d_in[3];
  const float* W_r3     = (const float*)d_in[4];
  const float* att_W    = (const float*)d_in[5];
  const float* att_b    = (const float*)d_in[6];
  const float* att_q    = (const float*)d_in[7];
  const float* bn_gamma = (const float*)d_in[8];
  const float* bn_beta  = (const float*)d_in[9];
  const int* src_r1 = (const int*)d_in[10];
  const int* dst_r1 = (const int*)d_in[11];
  const int* src_r2 = (const int*)d_in[12];
  const int* dst_r2 = (const int*)d_in[13];
  const int* src_r3 = (const int*)d_in[14];
  const int* dst_r3 = (const int*)d_in[15];

  const int NP = in_sizes[0] / 128;
  const int NA = in_sizes[1] / 128;
  const int E1 = in_sizes[10];
  const int E2 = in_sizes[12];
  const int E3 = in_sizes[14];

  // ---- workspace layout (floats); zero-initialized prefix first ----
  float* ws      = (float*)d_ws;
  float* degOut1 = ws;                        // NA  (authors, r1 src)
  float* degIn1  = degOut1 + NA;              // NP
  float* degOut2 = degIn1 + NP;               // NP
  float* degIn2  = degOut2 + NP;              // NP
  float* degOut3 = degIn2 + NP;               // NA
  float* degIn3  = degOut3 + NA;              // NA
  float* z1      = degIn3 + NA;               // NP*128 (agg author->paper)
  float* z2      = z1 + (size_t)NP * 128;     // NP*128 (agg paper->paper)
  float* z3      = z2 + (size_t)NP * 128;     // NA*128 (agg author->author)
  float* wAcc    = z3 + (size_t)NA * 128;     // 4
  float* statsP  = wAcc + 4;                  // 256 (sum,sumsq)
  float* statsA  = statsP + 256;              // 256
  // --- end of zeroed region ---
  float* betaV   = statsA + 256;              // 4
  float* cnP     = betaV + 4;                 // 256 (mu,invstd)
  float* cnA     = cnP + 256;                 // 256
  float* m1      = cnA + 256;                 // NA*128
  float* m2      = m1 + (size_t)NA * 128;     // NP*128
  float* m3      = m2 + (size_t)NP * 128;     // NA*128
  (void)m3; (void)ws_size; (void)n_in; (void)out_size;

  const size_t zeroBytes = (size_t)(statsA + 256 - ws) * sizeof(float);
  hipMemsetAsync(d_ws, 0, zeroBytes, stream);

  // degrees
  k_degrees<<<(E1 + 255) / 256, 256, 0, stream>>>(src_r1, dst_r1, degOut1, degIn1, E1);
  k_degrees<<<(E2 + 255) / 256, 256, 0, stream>>>(src_r2, dst_r2, degOut2, degIn2, E2);
  k_degrees<<<(E3 + 255) / 256, 256, 0, stream>>>(src_r3, dst_r3, degOut3, degIn3, E3);

  // m = (h * rsqrt(deg_out)) @ W   (WMMA f32, wave per 16x128 row block)
  const int gbA = (NA / 16 + 7) / 8;
  const int gbP = (NP / 16 + 7) / 8;
  k_gemm_scaled<<<gbA, 256, 0, stream>>>(h_author, degOut1, W_r1, m1, NA);
  k_gemm_scaled<<<gbP, 256, 0, stream>>>(h_paper,  degOut2, W_r2, m2, NP);
  k_gemm_scaled<<<gbA, 256, 0, stream>>>(h_author, degOut3, W_r3, m3, NA);

  // segment-sum over edges (wave per edge, float4 per lane)
  k_scatter<<<(E1 * 32 + 255) / 256, 256, 0, stream>>>(m1, src_r1, dst_r1, z1, E1);
  k_scatter<<<(E2 * 32 + 255) / 256, 256, 0, stream>>>(m2, src_r2, dst_r2, z2, E2);
  k_scatter<<<(E3 * 32 + 255) / 256, 256, 0, stream>>>(m3, src_r3, dst_r3, z3, E3);

  // z *= rsqrt(deg_in)
  k_scale_rows<<<(NP * 32 + 255) / 256, 256, 0, stream>>>(z1, degIn1, NP);
  k_scale_rows<<<(NP * 32 + 255) / 256, 256, 0, stream>>>(z2, degIn2, NP);
  k_scale_rows<<<(NA * 32 + 255) / 256, 256, 0, stream>>>(z3, degIn3, NA);

  // semantic attention logits for the two paper relations (authors: beta=1)
  k_attention<<<gbP, 256, 0, stream>>>(z1, att_W, att_b, att_q, wAcc + 0, NP);
  k_attention<<<gbP, 256, 0, stream>>>(z2, att_W, att_b, att_q, wAcc + 1, NP);
  k_softmax2<<<1, 32, 0, stream>>>(wAcc, betaV, 1.0f / (float)NP);

  // batchnorm column stats of combined outputs
  const int rowsPerBlock = 128;
  k_colstats<<<(NP + rowsPerBlock - 1) / rowsPerBlock, 128, 0, stream>>>(
      z1, z2, betaV, 0, 1, statsP, NP, rowsPerBlock);
  k_colstats<<<(NA + rowsPerBlock - 1) / rowsPerBlock, 128, 0, stream>>>(
      z3, z3, betaV, 2, 3, statsA, NA, rowsPerBlock);
  k_colnorm<<<1, 128, 0, stream>>>(statsP, cnP, 1.0f / (float)NP);
  k_colnorm<<<1, 128, 0, stream>>>(statsA, cnA, 1.0f / (float)NA);

  // fused combine + BN + row-L2, written straight to d_out
  float* out_p = (float*)d_out;
  float* out_a = out_p + (size_t)NP * 128;
  k_finalize<<<(NP * 32 + 255) / 256, 256, 0, stream>>>(z1, z2, betaV, 0, 1, cnP,
                                                        bn_gamma, bn_beta, out_p, NP);
  k_finalize<<<(NA * 32 + 255) / 256, 256, 0, stream>>>(z3, z3, betaV, 2, 3, cnA,
                                                        bn_gamma, bn_beta, out_a, NA);
}